// CRF_21431886807256
// MI455X (gfx1250) — compile-verified
//
#include <hip/hip_runtime.h>
#include <hip/hip_bf16.h>

typedef float v2f __attribute__((ext_vector_type(2)));
typedef float v8f __attribute__((ext_vector_type(8)));

#define WAVE 32

// ds_swizzle with embedded wait (for serial reduction hops).
// offset encoding (group-of-32): {xor_mask[14:10], or_mask[9:5]=0, and_mask[4:0]=0x1f}
#define SWZ_WAIT(dst, src, IMMSTR)                                                     \
    asm volatile("ds_swizzle_b32 %0, %1 offset:" IMMSTR "\n\ts_wait_dscnt 0x0"         \
                 : "=v"(dst) : "v"(src))

// ds_swizzle xor-16 (swap lane halves), no wait — batch these, then SWZ_FENCE.
#define SWZ16_NOWAIT(dst, src)                                                         \
    asm volatile("ds_swizzle_b32 %0, %1 offset:0x401f" : "=v"(dst) : "v"(src))

// One wait covering a batch of swizzles; ties the 8 results so consumers can't
// be scheduled before the wait.
#define SWZ_FENCE8(c0, c1, c2, c3, c4, c5, c6, c7)                                     \
    asm volatile("s_wait_dscnt 0x0"                                                    \
                 : "+v"(c0), "+v"(c1), "+v"(c2), "+v"(c3),                             \
                   "+v"(c4), "+v"(c5), "+v"(c6), "+v"(c7)                              \
                 :: "memory")

// A operand: 4 k-slices of a 16x16 f32 matrix in WMMA A layout.
// slice kb, component {x,y}: element (M = lane&15, K = 4*kb + {0,1} + 2*(lane>=16))
struct A4 { v2f k0, k1, k2, k3; };

// D = A * R, A 16x16 as 4 A-slices, R 16x16 in WMMA C/D layout:
// r[i] = element (M = i + 8*(lane>=16), N = lane&15).
// B slices (4x16: VGPR0 rows {4kb,4kb+2}, VGPR1 rows {4kb+1,4kb+3}) need only a
// lane-half swap of R: 8 ds_swizzle(xor16) issued together + 1 wait + 8 selects.
__device__ __forceinline__ v8f mul_AR(const A4& a, v8f r, int lane) {
    const bool lo = lane < 16;
    float c0, c1, c2, c3, c4, c5, c6, c7;   // c_i = r[i] with halves swapped
    SWZ16_NOWAIT(c0, r[0]); SWZ16_NOWAIT(c1, r[1]);
    SWZ16_NOWAIT(c2, r[2]); SWZ16_NOWAIT(c3, r[3]);
    SWZ16_NOWAIT(c4, r[4]); SWZ16_NOWAIT(c5, r[5]);
    SWZ16_NOWAIT(c6, r[6]); SWZ16_NOWAIT(c7, r[7]);
    SWZ_FENCE8(c0, c1, c2, c3, c4, c5, c6, c7);

    v2f b0, b1, b2, b3;
    b0.x = lo ? r[0] : c2;  b0.y = lo ? r[1] : c3;   // rows 0..3
    b1.x = lo ? r[4] : c6;  b1.y = lo ? r[5] : c7;   // rows 4..7
    b2.x = lo ? c0 : r[2];  b2.y = lo ? c1 : r[3];   // rows 8..11
    b3.x = lo ? c4 : r[6];  b3.y = lo ? c5 : r[7];   // rows 12..15

    v8f d = {0.f, 0.f, 0.f, 0.f, 0.f, 0.f, 0.f, 0.f};
    d = __builtin_amdgcn_wmma_f32_16x16x4_f32(false, a.k0, false, b0, (short)0, d, false, false);
    d = __builtin_amdgcn_wmma_f32_16x16x4_f32(false, a.k1, false, b1, (short)0, d, false, false);
    d = __builtin_amdgcn_wmma_f32_16x16x4_f32(false, a.k2, false, b2, (short)0, d, false, false);
    d = __builtin_amdgcn_wmma_f32_16x16x4_f32(false, a.k3, false, b3, (short)0, d, false, false);
    return d;
}

// Scale R by s ~ 1/max(R) (raw v_rcp is fine: ls -= log(s) compensates exactly
// for whatever scale was actually applied). Returns -log(s).
__device__ __forceinline__ float rescale(v8f& r, int lane) {
    float m = r[0];
#pragma unroll
    for (int i = 1; i < 8; ++i) m = fmaxf(m, r[i]);
    float t;
    SWZ_WAIT(t, m, "0x401f"); m = fmaxf(m, t);   // xor 16
    SWZ_WAIT(t, m, "0x201f"); m = fmaxf(m, t);   // xor 8
    SWZ_WAIT(t, m, "0x101f"); m = fmaxf(m, t);   // xor 4
    SWZ_WAIT(t, m, "0x081f"); m = fmaxf(m, t);   // xor 2
    SWZ_WAIT(t, m, "0x041f"); m = fmaxf(m, t);   // xor 1
    const float s = __builtin_amdgcn_rcpf(m);
#pragma unroll
    for (int i = 0; i < 8; ++i) r[i] *= s;
    return -__logf(s);
}

__device__ __forceinline__ v8f identity16(int lane) {
    const int  m  = lane & 15;
    const bool hi = lane >= 16;
    v8f r;
#pragma unroll
    for (int i = 0; i < 8; ++i) r[i] = (m == i + (hi ? 8 : 0)) ? 1.0f : 0.0f;
    return r;
}

// Load a 16x16 row-major matrix into WMMA A layout (8 scalar loads, L2-hit).
__device__ __forceinline__ A4 loadA(const float* __restrict__ q, int m, bool hi) {
    const int k0 = hi ? 2 : 0;
    A4 a;
    a.k0 = (v2f){q[m * 16 + 0  + k0], q[m * 16 + 1  + k0]};
    a.k1 = (v2f){q[m * 16 + 4  + k0], q[m * 16 + 5  + k0]};
    a.k2 = (v2f){q[m * 16 + 8  + k0], q[m * 16 + 9  + k0]};
    a.k3 = (v2f){q[m * 16 + 12 + k0], q[m * 16 + 13 + k0]};
    return a;
}

// Store R (C/D layout) as row-major 16x16 (coalesced: 2x64B per store).
__device__ __forceinline__ void storeR(float* __restrict__ q, v8f r, int m, bool hi) {
#pragma unroll
    for (int i = 0; i < 8; ++i) q[(i + (hi ? 8 : 0)) * 16 + m] = r[i];
}

// ---------------- Phase 1: per-(batch,chunk) transposed matrix products ----------------
// Q_chunk = (P_a * ... * P_b)^T via Q <- P_t^T * Q, t ascending.
// P_t^T[m][k] = exp(trans[k][m]) * exp(em[t][m]); P_0^T[m][k] = exp(start[m] + em[0][m]).
__global__ __launch_bounds__(WAVE)
void crf_phase1(const float* __restrict__ em, const float* __restrict__ trans,
                const float* __restrict__ startt,
                float* __restrict__ qws, float* __restrict__ lsws) {
    const int T = 2048, K = 16, CHUNK = 32, NC = 64;
    const int wg   = blockIdx.x;
    const int b    = wg / NC;
    const int c    = wg % NC;
    const int lane = threadIdx.x;
    const int m    = lane & 15;
    const bool hi  = lane >= 16;

    // Loop-invariant: exp of the 8 transition entries this lane contributes.
    float etK[8];
#pragma unroll
    for (int kb = 0; kb < 4; ++kb)
#pragma unroll
        for (int v = 0; v < 2; ++v)
            etK[kb * 2 + v] = __expf(trans[(4 * kb + v + (hi ? 2 : 0)) * K + m]);

    v8f r = identity16(lane);
    float ls = 0.f;
    const int t0 = c * CHUNK;

    for (int s = 0; s < CHUNK; ++s) {
        const int t = t0 + s;
        const float ee = __expf(em[(b * T + t) * K + m]);   // exp(em[t][m]), 1 exp/step
        A4 a;
        if (t == 0) {
            const float av = __expf(startt[m]) * ee;        // first matrix: rows all equal
            a.k0 = (v2f){av, av}; a.k1 = (v2f){av, av};
            a.k2 = (v2f){av, av}; a.k3 = (v2f){av, av};
        } else {
            a.k0 = (v2f){etK[0] * ee, etK[1] * ee};
            a.k1 = (v2f){etK[2] * ee, etK[3] * ee};
            a.k2 = (v2f){etK[4] * ee, etK[5] * ee};
            a.k3 = (v2f){etK[6] * ee, etK[7] * ee};
        }
        r = mul_AR(a, r, lane);
        if ((s & 3) == 3) ls += rescale(r, lane);           // growth <= ~2^52 per 4 steps
    }

    storeR(qws + (b * NC + c) * 256, r, m, hi);
    if (lane == 0) lsws[b * NC + c] = ls;
}

// ---------------- Phase 2a: combine 8 chunks per group ----------------
// S_g = Q_{8g+7} * ... * Q_{8g}
__global__ __launch_bounds__(WAVE)
void crf_phase2a(const float* __restrict__ qws, const float* __restrict__ lsws,
                 float* __restrict__ sws, float* __restrict__ ls2ws) {
    const int NC = 64, GRP = 8, NG = 8;
    const int wg   = blockIdx.x;           // 0..127
    const int b    = wg / NG;
    const int g    = wg % NG;
    const int lane = threadIdx.x;
    const int m    = lane & 15;
    const bool hi  = lane >= 16;

    v8f r = identity16(lane);
    float ls = 0.f;
    for (int j = 0; j < GRP; ++j) {
        const int c = g * GRP + j;
        A4 a = loadA(qws + (b * NC + c) * 256, m, hi);
        r = mul_AR(a, r, lane);
        if ((j & 3) == 3) ls += rescale(r, lane);           // inputs <=1: growth <=16^4
        ls += lsws[b * NC + c];
    }
    storeR(sws + (b * NG + g) * 256, r, m, hi);
    if (lane == 0) ls2ws[b * NG + g] = ls;
}

// ---------------- Phase 2b: combine 8 groups per batch; emit logZ ----------------
// Qtot = S_7 * ... * S_0 = (P_0 * ... * P_{T-1})^T
__global__ __launch_bounds__(WAVE)
void crf_phase2b(const float* __restrict__ sws, const float* __restrict__ ls2ws,
                 const float* __restrict__ endt, float* __restrict__ logZ) {
    const int NG = 8;
    const int b    = blockIdx.x;
    const int lane = threadIdx.x;
    const int m    = lane & 15;
    const bool hi  = lane >= 16;

    v8f r = identity16(lane);
    float ls = 0.f;
    for (int g = 0; g < NG; ++g) {
        A4 a = loadA(sws + (b * NG + g) * 256, m, hi);
        r = mul_AR(a, r, lane);
        if ((g & 3) == 3) ls += rescale(r, lane);
        ls += ls2ws[b * NG + g];
    }

    // logZ_b = log( sum_j Qtot[j][0] * exp(end[j]) ) + ls.
    // Column N=0 of Qtot: lane 0 (rows 0-7), lane 16 (rows 8-15).
    float p = 0.f;
    if (m == 0) {
#pragma unroll
        for (int i = 0; i < 8; ++i) p += r[i] * __expf(endt[i + (hi ? 8 : 0)]);
    }
    float ph;
    SWZ_WAIT(ph, p, "0x401f");      // xor 16: lane0 gets lane16's half
    p += ph;
    if (lane == 0) logZ[b] = __logf(p) + ls;
}

// ---------------- Tag-sequence score ----------------
__global__ __launch_bounds__(256)
void crf_seqscore(const float* __restrict__ em, const long long* __restrict__ tags,
                  const float* __restrict__ trans, const float* __restrict__ startt,
                  const float* __restrict__ endt, float* __restrict__ seq) {
    const int T = 2048, K = 16;
    const int b = blockIdx.x, tid = threadIdx.x;
    float s = 0.f;
    for (int t = tid; t < T; t += 256) {
        const int tg = (int)tags[b * T + t];
        s += em[(b * T + t) * K + tg];
        if (t > 0)      s += trans[(int)tags[b * T + t - 1] * K + tg];
        if (t == 0)     s += startt[tg];
        if (t == T - 1) s += endt[tg];
    }
    __shared__ float red[256];
    red[tid] = s;
    __syncthreads();
#pragma unroll
    for (int off = 128; off >= 1; off >>= 1) {
        if (tid < off) red[tid] += red[tid + off];
        __syncthreads();
    }
    if (tid == 0) seq[b] = red[0];
}

// ---------------- Finalize: mean over batch ----------------
__global__ __launch_bounds__(WAVE)
void crf_finalize(const float* __restrict__ logZ, const float* __restrict__ seq,
                  float* __restrict__ out) {
    const int lane = threadIdx.x;
    float d = (lane < 16) ? (logZ[lane] - seq[lane]) : 0.f;
    float t;
    SWZ_WAIT(t, d, "0x201f"); d += t;   // xor 8
    SWZ_WAIT(t, d, "0x101f"); d += t;   // xor 4
    SWZ_WAIT(t, d, "0x081f"); d += t;   // xor 2
    SWZ_WAIT(t, d, "0x041f"); d += t;   // xor 1
    if (lane == 0) out[0] = d / 16.0f;
}

extern "C" void kernel_launch(void* const* d_in, const int* in_sizes, int n_in,
                              void* d_out, int out_size, void* d_ws, size_t ws_size,
                              hipStream_t stream) {
    const float*     em     = (const float*)d_in[0];      // (16, 2048, 16) f32
    const long long* tags   = (const long long*)d_in[1];  // (16, 2048) i64
    const float*     trans  = (const float*)d_in[2];      // (16, 16) f32
    const float*     startt = (const float*)d_in[3];      // (16,) f32
    const float*     endt   = (const float*)d_in[4];      // (16,) f32
    float*           out    = (float*)d_out;

    float* w     = (float*)d_ws;
    float* qws   = w;                       // 16*64*256 = 262144 floats
    float* lsws  = qws  + 16 * 64 * 256;    // 1024 floats
    float* sws   = lsws + 1024;             // 16*8*256 = 32768 floats
    float* ls2ws = sws  + 16 * 8 * 256;     // 128 floats
    float* logZ  = ls2ws + 128;             // 16 floats
    float* seq   = logZ + 16;               // 16 floats

    crf_phase1  <<<dim3(16 * 64), dim3(WAVE), 0, stream>>>(em, trans, startt, qws, lsws);
    crf_seqscore<<<dim3(16),      dim3(256),  0, stream>>>(em, tags, trans, startt, endt, seq);
    crf_phase2a <<<dim3(16 * 8),  dim3(WAVE), 0, stream>>>(qws, lsws, sws, ls2ws);
    crf_phase2b <<<dim3(16),      dim3(WAVE), 0, stream>>>(sws, ls2ws, endt, logZ);
    crf_finalize<<<dim3(1),       dim3(WAVE), 0, stream>>>(logZ, seq, out);
}